// BoundaryLoss_84688165143115
// MI455X (gfx1250) — compile-verified
//
#include <hip/hip_runtime.h>
#include <math.h>

// BoundaryLoss on MI455X (gfx1250, wave32).
// inputs: d_in[0]=logits f32 [16,1,512,512], d_in[1]=targets f32 [16,1,512,512]
// output: d_out[0] = mean(clip(sigmoid(logits)) * (edt(t)+edt(~t)))  (f32 scalar)
// scratch layout in d_ws: g2t[N] | g2n[N] | partials[B*H]   (32MB + 32KB)

#define BATCH 16
#define HH    512
#define WW    512
#define NPIX  (BATCH * HH * WW)
#define NROWS (BATCH * HH)       // 8192
#define NCOLS (BATCH * WW)       // 8192
#define BIGD  1.0e9f

typedef float v2f __attribute__((ext_vector_type(2)));
typedef float v8f __attribute__((ext_vector_type(8)));

// Exact f32 wave(32) sum via V_WMMA_F32_16X16X4_F32.
// A(16x4): lanes 0-15 hold (M=lane, K=0..1), lanes 16-31 hold (M=lane-16, K=2..3).
// With B = all-ones, D[m,n] = sum_k A[m,k] (independent of n).
// C/D(16x16): lane L<16 vgpr v -> (M=v, N=L); lane L>=16 vgpr v -> (M=v+8, N=L-16).
// So sum_v c[v] on lane0 = sum_{m=0..7} rowsum_m, on lane16 = sum_{m=8..15} rowsum_m.
__device__ __forceinline__ float wave_sum_wmma(float val) {
    v2f a; a[0] = val;  a[1] = 0.0f;
    v2f b; b[0] = 1.0f; b[1] = 1.0f;
    v8f c = {};
    c = __builtin_amdgcn_wmma_f32_16x16x4_f32(
            /*neg_a=*/false, a, /*neg_b=*/false, b,
            /*c_mod=*/(short)0, c, /*reuse_a=*/false, /*reuse_b=*/false);
    float s = c[0] + c[1] + c[2] + c[3] + c[4] + c[5] + c[6] + c[7];
    return __shfl(s, 0, 32) + __shfl(s, 16, 32);
}

// Same primitive but accumulating a stream of lane-pairs into C across iterations.
__device__ __forceinline__ v8f wmma_acc_pair(v8f c, float x0, float x1) {
    v2f a; a[0] = x0;   a[1] = x1;
    v2f b; b[0] = 1.0f; b[1] = 1.0f;
    return __builtin_amdgcn_wmma_f32_16x16x4_f32(
            false, a, false, b, (short)0, c, false, false);
}

// ---------------------------------------------------------------------------
// Kernel 1: exact 1D distance along H for both t and ~t, per column (b,w).
// Forward scan writes running distances; backward scan combines, squares,
// and stores g2 in place. Adjacent threads = adjacent w => fully coalesced.
// ---------------------------------------------------------------------------
__global__ __launch_bounds__(256) void col_pass(const float* __restrict__ targets,
                                                float* __restrict__ g2t,
                                                float* __restrict__ g2n) {
    int tid = blockIdx.x * blockDim.x + threadIdx.x;     // 0 .. NCOLS-1
    int b = tid / WW, w = tid % WW;
    size_t base = (size_t)b * HH * WW + (size_t)w;

    float ct = BIGD, cn = BIGD;
    for (int h = 0; h < HH; ++h) {
        size_t idx = base + (size_t)h * WW;
        __builtin_prefetch(&targets[idx + (size_t)16 * WW], 0, 0); // global_prefetch_b8
        float tv = targets[idx];
        bool  tt = tv > 0.5f;
        ct = fminf(ct + 1.0f, tt ? 0.0f : BIGD);
        cn = fminf(cn + 1.0f, tt ? BIGD : 0.0f);
        g2t[idx] = ct;
        g2n[idx] = cn;
    }
    ct = BIGD; cn = BIGD;
    for (int h = HH - 1; h >= 0; --h) {
        size_t idx = base + (size_t)h * WW;
        float tv = targets[idx];
        bool  tt = tv > 0.5f;
        ct = fminf(ct + 1.0f, tt ? 0.0f : BIGD);
        cn = fminf(cn + 1.0f, tt ? BIGD : 0.0f);
        float dt = fminf(g2t[idx], ct);
        float dn = fminf(g2n[idx], cn);
        g2t[idx] = dt * dt;
        g2n[idx] = dn * dn;
    }
}

// ---------------------------------------------------------------------------
// Kernel 2: per-row exact min-plus transform (outward scan w/ early exit),
// fused with sigmoid(logit)*dt and a WMMA wave reduction -> per-row partial.
// One block (512 threads, 16 waves) per row; g2 rows staged in LDS.
// Early exit is exact: g2>=0 and d^2 exactly representable => candidates with
// d^2 >= best cannot lower the min.
// ---------------------------------------------------------------------------
__global__ __launch_bounds__(512) void row_pass(const float* __restrict__ logits,
                                                const float* __restrict__ g2t,
                                                const float* __restrict__ g2n,
                                                float* __restrict__ partials) {
    __shared__ float st[WW];
    __shared__ float sn[WW];
    __shared__ float wsum[16];

    int row = blockIdx.x;            // b*H + h
    int j   = threadIdx.x;           // 0..511
    size_t base = (size_t)row * WW;

    st[j] = g2t[base + j];
    sn[j] = g2n[base + j];
    __syncthreads();

    float bt = st[j];
    float bn = sn[j];
    for (int d = 1; d < WW; ++d) {
        float d2 = (float)(d * d);
        if (d2 >= bt && d2 >= bn) break;
        int jl = j - d, jr = j + d;
        if (jl >= 0) {
            bt = fminf(bt, st[jl] + d2);
            bn = fminf(bn, sn[jl] + d2);
        }
        if (jr < WW) {
            bt = fminf(bt, st[jr] + d2);
            bn = fminf(bn, sn[jr] + d2);
        }
    }
    float dtv = sqrtf(bt) + sqrtf(bn);           // edt(t) + edt(~t) at this pixel

    float x = logits[base + j];
    float p = 1.0f / (1.0f + expf(-x));
    p = fminf(fmaxf(p, 1e-7f), 1.0f - 1e-7f);
    float val = p * dtv;

    // EXEC is all-ones here (all 512 threads live, loop reconverged).
    float ws = wave_sum_wmma(val);
    int lane = threadIdx.x & 31;
    int wave = threadIdx.x >> 5;
    if (lane == 0) wsum[wave] = ws;
    __syncthreads();
    if (threadIdx.x == 0) {
        float s = 0.0f;
        #pragma unroll
        for (int i = 0; i < 16; ++i) s += wsum[i];  // fixed order: deterministic
        partials[row] = s;
    }
}

// ---------------------------------------------------------------------------
// Kernel 3: single-block deterministic reduction of NROWS partials with
// WMMA accumulation in the loop; writes the mean.
// 256 threads = 8 waves; each wave consumes 1024 partials (16 x 64/iter).
// ---------------------------------------------------------------------------
__global__ __launch_bounds__(256) void final_reduce(const float* __restrict__ partials,
                                                    float* __restrict__ out) {
    __shared__ float wsum[8];
    int lane = threadIdx.x & 31;
    int wave = threadIdx.x >> 5;

    v8f c = {};
    int base = wave * (NROWS / 8);               // 1024 per wave
    #pragma unroll 4
    for (int it = 0; it < (NROWS / 8) / 64; ++it) {   // 16 iterations
        int idx = base + it * 64 + lane * 2;
        c = wmma_acc_pair(c, partials[idx], partials[idx + 1]);
    }
    float s = c[0] + c[1] + c[2] + c[3] + c[4] + c[5] + c[6] + c[7];
    float wtot = __shfl(s, 0, 32) + __shfl(s, 16, 32);
    if (lane == 0) wsum[wave] = wtot;
    __syncthreads();
    if (threadIdx.x == 0) {
        float g = 0.0f;
        #pragma unroll
        for (int i = 0; i < 8; ++i) g += wsum[i];
        out[0] = g / (float)NPIX;
    }
}

extern "C" void kernel_launch(void* const* d_in, const int* in_sizes, int n_in,
                              void* d_out, int out_size, void* d_ws, size_t ws_size,
                              hipStream_t stream) {
    const float* logits  = (const float*)d_in[0];
    const float* targets = (const float*)d_in[1];

    float* g2t      = (float*)d_ws;          // NPIX floats
    float* g2n      = g2t + NPIX;            // NPIX floats
    float* partials = g2n + NPIX;            // NROWS floats

    col_pass<<<NCOLS / 256, 256, 0, stream>>>(targets, g2t, g2n);
    row_pass<<<NROWS, WW, 0, stream>>>(logits, g2t, g2n, partials);
    final_reduce<<<1, 256, 0, stream>>>(partials, (float*)d_out);
}